// DepthToAttribute_81037442941098
// MI455X (gfx1250) — compile-verified
//
#include <hip/hip_runtime.h>
#include <hip/hip_bf16.h>
#include <math.h>

// ---------------------------------------------------------------------------
// Problem constants (from reference)
// ---------------------------------------------------------------------------
#define FARP   10.0f
#define NEARP  0.1f
#define EPSN   1e-12f

constexpr int CAMH = 768;
constexpr int CAMW = 1024;
constexpr int PRJH = 800;
constexpr int PRJW = 1280;
constexpr int NB   = 8;
constexpr int HWc  = CAMH * CAMW;      // 786432 (divisible by 256)

typedef __attribute__((ext_vector_type(2))) float        v2f;
typedef __attribute__((ext_vector_type(8))) float        v8f;
typedef __attribute__((ext_vector_type(4))) unsigned int v4u;
typedef __attribute__((ext_vector_type(4))) int          v4i;
typedef __attribute__((ext_vector_type(8))) int          v8i;

__device__ __forceinline__ float sigm100(float x) {
    return 1.0f / (1.0f + expf(-100.0f * x));
}
__device__ __forceinline__ float clamp01(float x) {
    return fminf(fmaxf(x, 0.0f), 1.0f);
}

// B-matrix entry for the fused transform:
//   cols 0..2 : [R^T ; t]      -> prj_pts3d = [p,1] @ B
//   cols 3..5 : [(K R)^T; K t] -> uvw       = [p,1] @ B
__device__ __forceinline__ float bmat_entry(int k, int n,
                                            const float* __restrict__ RT,
                                            const float* __restrict__ Km) {
    if (n < 3) {
        return (k < 3) ? RT[n * 4 + k] : RT[n * 4 + 3];
    } else if (n < 6) {
        int j = n - 3;
        if (k < 3)
            return Km[j * 3 + 0] * RT[0 * 4 + k] +
                   Km[j * 3 + 1] * RT[1 * 4 + k] +
                   Km[j * 3 + 2] * RT[2 * 4 + k];
        return Km[j * 3 + 0] * RT[3] + Km[j * 3 + 1] * RT[7] + Km[j * 3 + 2] * RT[11];
    }
    return 0.0f;
}

// ---------------------------------------------------------------------------
// K1: geometry. Per pixel: d=1/clip(depth), p = npix*d, then per-16-pixel
// WMMA computes [prj_pts3d | uvw]. Emits v_wmma_f32_16x16x4_f32.
// Outputs: grid (out buffer, (H,W,2)) and xyz planes (3,H,W) in ws.
// ---------------------------------------------------------------------------
__global__ __launch_bounds__(256) void geom_kernel(
    const float* __restrict__ depth, const float* __restrict__ npx3,
    const float* __restrict__ RT,    const float* __restrict__ Km,
    float* __restrict__ grid_out,    float* __restrict__ xyz) {
    const int lane = threadIdx.x & 31;
    const int wave = threadIdx.x >> 5;
    const int pix  = blockIdx.x * 256 + wave * 32 + lane;   // always in-range

    float dd   = fminf(fmaxf(depth[pix], 1.0f / FARP), 1.0f / NEARP);
    float dinv = 1.0f / dd;
    float px = npx3[pix * 3 + 0] * dinv;
    float py = npx3[pix * 3 + 1] * dinv;
    float pz = npx3[pix * 3 + 2] * dinv;

    // B operand (shared by both WMMAs of the wave): 4x16, f32 layout:
    // lanes 0-15 hold rows K=0,1; lanes 16-31 hold rows K=2,3.
    const int col = lane & 15;
    const int r0  = (lane < 16) ? 0 : 2;
    v2f bv;
    bv.x = bmat_entry(r0,     col, RT, Km);
    bv.y = bmat_entry(r0 + 1, col, RT, Km);

    // A operand, tile 0 (pixels 0..15 of the wave):
    // lanes 0-15: {A[m][0],A[m][1]} = {px,py}; lanes 16-31: {A[m][2],A[m][3]} = {pz,1}
    float z_lo = __shfl(pz, lane & 15, 32);
    v2f a0;
    a0.x = (lane < 16) ? px : z_lo;
    a0.y = (lane < 16) ? py : 1.0f;

    // A operand, tile 1 (pixels 16..31 of the wave)
    float x_hi = __shfl(px, (lane & 15) + 16, 32);
    float y_hi = __shfl(py, (lane & 15) + 16, 32);
    v2f a1;
    a1.x = (lane < 16) ? x_hi : pz;
    a1.y = (lane < 16) ? y_hi : 1.0f;

    v8f acc0 = {0.f, 0.f, 0.f, 0.f, 0.f, 0.f, 0.f, 0.f};
    v8f acc1 = {0.f, 0.f, 0.f, 0.f, 0.f, 0.f, 0.f, 0.f};
    acc0 = __builtin_amdgcn_wmma_f32_16x16x4_f32(false, a0, false, bv,
                                                 (short)0, acc0, false, false);
    acc1 = __builtin_amdgcn_wmma_f32_16x16x4_f32(false, a1, false, bv,
                                                 (short)0, acc1, false, false);

    // D layout -> per-pixel via LDS transpose (each wave owns its region).
    __shared__ float tileD[8][2][256];
    const int Nn    = lane & 15;
    const int Mbase = (lane < 16) ? 0 : 8;
#pragma unroll
    for (int r = 0; r < 8; ++r) {
        tileD[wave][0][(Mbase + r) * 16 + Nn] = acc0[r];
        tileD[wave][1][(Mbase + r) * 16 + Nn] = acc1[r];
    }
    __syncthreads();

    const int tsel = lane >> 4;
    const int m    = lane & 15;
    const float* drow = &tileD[wave][tsel][m * 16];
    float prjz = drow[2];
    float uu = drow[3], vv = drow[4], ww = drow[5];

    float gx = (uu / ww) * (2.0f / (float)PRJW) - 1.0f;
    float gy = (vv / ww) * (2.0f / (float)PRJH) - 1.0f;
    grid_out[pix * 2 + 0] = gx;
    grid_out[pix * 2 + 1] = gy;

    // xyz = [xy.x, xy.y, z]
    xyz[pix]           = (gx + 1.0f) * (float)(PRJW - 1) * 0.5f;
    xyz[HWc + pix]     = (gy + 1.0f) * (float)(PRJH - 1) * 0.5f;
    xyz[2 * HWc + pix] = prjz;
}

// ---------------------------------------------------------------------------
// K2: bilinear sample xyz at cam_rect_grid (align_corners=True, zero-pad).
// ---------------------------------------------------------------------------
__global__ __launch_bounds__(256) void rect_sample_kernel(
    const float* __restrict__ xyz, const float* __restrict__ rgrid,
    float* __restrict__ xyzr) {
    int pix = blockIdx.x * 256 + threadIdx.x;
    float gx = rgrid[pix * 2 + 0], gy = rgrid[pix * 2 + 1];
    float ix = (gx + 1.0f) * 0.5f * (float)(CAMW - 1);
    float iy = (gy + 1.0f) * 0.5f * (float)(CAMH - 1);
    float x0f = floorf(ix), y0f = floorf(iy);
    float wx = ix - x0f, wy = iy - y0f;
    int x0 = (int)x0f, y0 = (int)y0f;
    float o0 = 0.f, o1 = 0.f, o2 = 0.f;
#pragma unroll
    for (int dy = 0; dy < 2; ++dy)
#pragma unroll
        for (int dx = 0; dx < 2; ++dx) {
            int xi = x0 + dx, yi = y0 + dy;
            float w = (dx ? wx : 1.0f - wx) * (dy ? wy : 1.0f - wy);
            float vmask = ((xi >= 0) && (xi < CAMW) && (yi >= 0) && (yi < CAMH)) ? 1.0f : 0.0f;
            int xc = min(max(xi, 0), CAMW - 1);
            int yc = min(max(yi, 0), CAMH - 1);
            int off = yc * CAMW + xc;
            float f = w * vmask;
            o0 += f * xyz[off];
            o1 += f * xyz[HWc + off];
            o2 += f * xyz[2 * HWc + off];
        }
    xyzr[pix] = o0;
    xyzr[HWc + pix] = o1;
    xyzr[2 * HWc + pix] = o2;
}

// ---------------------------------------------------------------------------
// K3: per-row bitonic sort of key = 1e4*round(x)+10*round(y)+z over W=1024.
// The 3-plane row segment of xyz_r (12 KB) is DMA-staged into LDS with one
// TENSOR_LOAD_TO_LDS (2-D tile: 1024 x 3, dim0 stride = H*W), so both the key
// build and the random post-sort diff gathers hit LDS instead of HBM.
// One 1024-thread workgroup (32 wave32) per row.
// ---------------------------------------------------------------------------
__global__ __launch_bounds__(1024) void sort_mask_kernel(
    const float* __restrict__ xyzr, const float* __restrict__ RT,
    float* __restrict__ maskr) {
    __shared__ float srow[3][1024];     // TDM destination (xyz_r row, 3 planes)
    __shared__ float skey[1024];
    __shared__ int   sidx[1024];
    const int row  = blockIdx.x;
    const int tid  = threadIdx.x;
    const int base = row * CAMW;

    if (tid < 32) {   // single wave issues the TDM op (TDM ignores EXEC)
        unsigned long long gaddr = (unsigned long long)(const void*)(xyzr + base);
        unsigned lds_addr = (unsigned)(unsigned long long)(const void*)&srow[0][0];

        // D# group 0: count=1 | lds_addr | global_addr | type=2
        v4u g0;
        g0[0] = 1u;                                               // count = 1
        g0[1] = lds_addr;                                         // LDS byte addr
        g0[2] = (unsigned)(gaddr & 0xFFFFFFFFull);                // gaddr[31:0]
        g0[3] = (unsigned)((gaddr >> 32) & 0x01FFFFFFull)         // gaddr[56:32]
              | (2u << 30);                                       // type = image
        // D# group 1: data_size=4B, tensor 1024x3, tile 1024x3, stride=H*W
        v8i g1;
        g1[0] = (int)(2u << 16);          // data_size = 2 (4 bytes)
        g1[1] = (int)(1024u << 16);       // tensor_dim0[15:0] = 1024
        g1[2] = (int)(3u << 16);          // tensor_dim0 hi=0, tensor_dim1 = 3
        g1[3] = (int)(1024u << 16);       // tensor_dim1 hi=0, tile_dim0 = 1024
        g1[4] = (int)3;                   // tile_dim1 = 3, tile_dim2 = 0
        g1[5] = (int)HWc;                 // tensor_dim0_stride[31:0] = 786432
        g1[6] = 0;                        // stride hi = 0, dim1_stride lo = 0
        g1[7] = 0;
        v4i gz  = {0, 0, 0, 0};                   // groups 2/3 unused (2-D)
        v8i gz8 = {0, 0, 0, 0, 0, 0, 0, 0};       // extra group (6-arg form)

        __builtin_amdgcn_tensor_load_to_lds(g0, g1, gz, gz, gz8, 0);
        __builtin_amdgcn_s_wait_tensorcnt(0);   // s_wait_tensorcnt 0
    }
    __syncthreads();

    float xr = srow[0][tid];
    float yr = srow[1][tid];
    float zr = srow[2][tid];
    skey[tid] = 10000.0f * rintf(xr) + 10.0f * rintf(yr) + zr;
    sidx[tid] = tid;
    __syncthreads();

    for (int k = 2; k <= 1024; k <<= 1) {
        for (int j = k >> 1; j > 0; j >>= 1) {
            int ixj = tid ^ j;
            if (ixj > tid) {
                bool up = ((tid & k) == 0);
                float a = skey[tid], b = skey[ixj];
                if ((a > b) == up) {
                    skey[tid] = b; skey[ixj] = a;
                    int t = sidx[tid]; sidx[tid] = sidx[ixj]; sidx[ixj] = t;
                }
            }
            __syncthreads();
        }
    }

    int icur = sidx[tid];
    float dx = 0.f, dyv = 0.f, dz = 0.f;
    if (tid > 0) {
        int iprev = sidx[tid - 1];
        dx  = srow[0][icur] - srow[0][iprev];
        dyv = srow[1][icur] - srow[1][iprev];
        dz  = srow[2][icur] - srow[2][iprev];
    }
    const float t_z = 0.001f;
    float occ_xy = sigm100(1.0f - fabsf(dx) - fabsf(dyv));
    float m_pos = 1.0f - occ_xy * sigm100(dz - t_z);
    float m_neg = 1.0f - occ_xy * sigm100(t_z - dz);
    float tx = RT[3];   // t[0]
    maskr[base + icur] = (tx > 0.0f) ? m_pos : m_neg;
}

// ---------------------------------------------------------------------------
// K4: sample mask_rect at cam_unrect_grid (align=True) -> occ_mask, and the
// "ones" footprint sample at grid (align=True over PHxPW) -> dl_mask.
// ---------------------------------------------------------------------------
__global__ __launch_bounds__(256) void unrect_kernel(
    const float* __restrict__ maskr, const float* __restrict__ ugrid,
    const float* __restrict__ grid_in, float* __restrict__ dl_out) {
    int pix = blockIdx.x * 256 + threadIdx.x;

    // mask sample (source H x W)
    float gx = ugrid[pix * 2 + 0], gy = ugrid[pix * 2 + 1];
    float ix = (gx + 1.0f) * 0.5f * (float)(CAMW - 1);
    float iy = (gy + 1.0f) * 0.5f * (float)(CAMH - 1);
    float x0f = floorf(ix), y0f = floorf(iy);
    float wx = ix - x0f, wy = iy - y0f;
    int x0 = (int)x0f, y0 = (int)y0f;
    float mval = 0.f;
#pragma unroll
    for (int dy = 0; dy < 2; ++dy)
#pragma unroll
        for (int dx = 0; dx < 2; ++dx) {
            int xi = x0 + dx, yi = y0 + dy;
            float w = (dx ? wx : 1.0f - wx) * (dy ? wy : 1.0f - wy);
            float vmask = ((xi >= 0) && (xi < CAMW) && (yi >= 0) && (yi < CAMH)) ? 1.0f : 0.0f;
            int xc = min(max(xi, 0), CAMW - 1);
            int yc = min(max(yi, 0), CAMH - 1);
            mval += w * vmask * maskr[yc * CAMW + xc];
        }
    float occ = sigm100(mval - 0.5f);

    // ones sample (source PH x PW), align=True: sum of in-bounds tap weights
    float ggx = grid_in[pix * 2 + 0], ggy = grid_in[pix * 2 + 1];
    float jx = (ggx + 1.0f) * 0.5f * (float)(PRJW - 1);
    float jy = (ggy + 1.0f) * 0.5f * (float)(PRJH - 1);
    float u0f = floorf(jx), v0f = floorf(jy);
    float uw = jx - u0f, vw = jy - v0f;
    int u0 = (int)u0f, v0 = (int)v0f;
    float ones = 0.f;
#pragma unroll
    for (int dy = 0; dy < 2; ++dy)
#pragma unroll
        for (int dx = 0; dx < 2; ++dx) {
            int xi = u0 + dx, yi = v0 + dy;
            float w = (dx ? uw : 1.0f - uw) * (dy ? vw : 1.0f - vw);
            float vmask = ((xi >= 0) && (xi < PRJW) && (yi >= 0) && (yi < PRJH)) ? 1.0f : 0.0f;
            ones += w * vmask;
        }
    dl_out[pix] = ones * occ;
}

// ---------------------------------------------------------------------------
// K5: Sobel normals on cam_pts3d (recomputed from depth), edge padding.
// Writes n_buf (3,H,W) and the broadcast output n (B,3,H,W).
// ---------------------------------------------------------------------------
__global__ __launch_bounds__(256) void normals_kernel(
    const float* __restrict__ depth, const float* __restrict__ npx3,
    float* __restrict__ nbuf, float* __restrict__ nout) {
    int pix = blockIdx.x * 256 + threadIdx.x;
    int h = pix / CAMW, w = pix % CAMW;

    const float SX[3][3] = {{-0.125f, 0.f, 0.125f},
                            {-0.25f,  0.f, 0.25f},
                            {-0.125f, 0.f, 0.125f}};
    const float SY[3][3] = {{-0.125f, -0.25f, -0.125f},
                            { 0.f,     0.f,    0.f},
                            { 0.125f,  0.25f,  0.125f}};
    float gx0 = 0.f, gx1 = 0.f, gx2 = 0.f;
    float gy0 = 0.f, gy1 = 0.f, gy2 = 0.f;
#pragma unroll
    for (int ky = 0; ky < 3; ++ky)
#pragma unroll
        for (int kx = 0; kx < 3; ++kx) {
            int yy = min(max(h + ky - 1, 0), CAMH - 1);
            int xx = min(max(w + kx - 1, 0), CAMW - 1);
            int off = yy * CAMW + xx;
            float dd = fminf(fmaxf(depth[off], 1.0f / FARP), 1.0f / NEARP);
            float dinv = 1.0f / dd;
            float p0 = npx3[off * 3 + 0] * dinv;
            float p1 = npx3[off * 3 + 1] * dinv;
            float p2 = npx3[off * 3 + 2] * dinv;
            float cx = SX[ky][kx], cy = SY[ky][kx];
            gx0 += p0 * cx; gx1 += p1 * cx; gx2 += p2 * cx;
            gy0 += p0 * cy; gy1 += p1 * cy; gy2 += p2 * cy;
        }
    // cross(dzdx, dzdy)
    float cx = gx1 * gy2 - gx2 * gy1;
    float cy = gx2 * gy0 - gx0 * gy2;
    float cz = gx0 * gy1 - gx1 * gy0;
    float nr = fmaxf(sqrtf(cx * cx + cy * cy + cz * cz), EPSN);
    float nx = -cx / nr, ny = -cy / nr, nz = -cz / nr;
    nbuf[pix] = nx; nbuf[HWc + pix] = ny; nbuf[2 * HWc + pix] = nz;
#pragma unroll
    for (int b = 0; b < NB; ++b) {
        nout[(b * 3 + 0) * HWc + pix] = nx;
        nout[(b * 3 + 1) * HWc + pix] = ny;
        nout[(b * 3 + 2) * HWc + pix] = nz;
    }
}

// ---------------------------------------------------------------------------
// K6: shading. Per pixel: lighting geometry once, then per batch: bilinear
// sample of Ip (align=False, zero-pad), Ip_w / Ic_diff / Ic_spec.
// ---------------------------------------------------------------------------
__global__ __launch_bounds__(256) void shade_kernel(
    const float* __restrict__ Ip,    const float* __restrict__ s,
    const float* __restrict__ depth, const float* __restrict__ npx3,
    const float* __restrict__ org,   const float* __restrict__ grid_in,
    const float* __restrict__ dl_in, const float* __restrict__ nbuf,
    float* __restrict__ o_ipw, float* __restrict__ o_diff,
    float* __restrict__ o_spec) {
    int pix = blockIdx.x * 256 + threadIdx.x;

    float gx = grid_in[pix * 2 + 0], gy = grid_in[pix * 2 + 1];
    float dl = dl_in[pix];

    float dd = fminf(fmaxf(depth[pix], 1.0f / FARP), 1.0f / NEARP);
    float dinv = 1.0f / dd;
    float px = npx3[pix * 3 + 0] * dinv;
    float py = npx3[pix * 3 + 1] * dinv;
    float pz = npx3[pix * 3 + 2] * dinv;

    float lx = org[0] - px, ly = org[1] - py, lz = org[2] - pz;
    float ln = fmaxf(sqrtf(lx * lx + ly * ly + lz * lz), EPSN);
    lx = lx / ln * dl; ly = ly / ln * dl; lz = lz / ln * dl;

    float pn = fmaxf(sqrtf(px * px + py * py + pz * pz), EPSN);
    float vx = -px / pn, vy = -py / pn, vz = -pz / pn;

    float nx = nbuf[pix], ny = nbuf[HWc + pix], nz = nbuf[2 * HWc + pix];
    float ndl = nx * lx + ny * ly + nz * lz;

    float hx = lx + vx, hy = ly + vy, hz = lz + vz;
    float hn = fmaxf(sqrtf(hx * hx + hy * hy + hz * hz), EPSN);
    hx /= hn; hy /= hn; hz /= hn;
    float rdv = hx * nx + hy * ny + hz * nz;

    // Ip bilinear taps (align_corners=False)
    float ix = ((gx + 1.0f) * (float)PRJW - 1.0f) * 0.5f;
    float iy = ((gy + 1.0f) * (float)PRJH - 1.0f) * 0.5f;
    float x0f = floorf(ix), y0f = floorf(iy);
    float wx = ix - x0f, wy = iy - y0f;
    int x0 = (int)x0f, y0 = (int)y0f, x1 = x0 + 1, y1 = y0 + 1;
    bool bx0 = (x0 >= 0) && (x0 < PRJW), bx1 = (x1 >= 0) && (x1 < PRJW);
    bool by0 = (y0 >= 0) && (y0 < PRJH), by1 = (y1 >= 0) && (y1 < PRJH);
    int xc0 = min(max(x0, 0), PRJW - 1), xc1 = min(max(x1, 0), PRJW - 1);
    int yc0 = min(max(y0, 0), PRJH - 1), yc1 = min(max(y1, 0), PRJH - 1);
    float f00 = (bx0 && by0) ? (1.0f - wx) * (1.0f - wy) : 0.0f;
    float f01 = (bx1 && by0) ? wx * (1.0f - wy) : 0.0f;
    float f10 = (bx0 && by1) ? (1.0f - wx) * wy : 0.0f;
    float f11 = (bx1 && by1) ? wx * wy : 0.0f;
    int o00 = yc0 * PRJW + xc0, o01 = yc0 * PRJW + xc1;
    int o10 = yc1 * PRJW + xc0, o11 = yc1 * PRJW + xc1;

    for (int b = 0; b < NB; ++b) {
        int bc = b * 3;
        if (b + 1 < NB)   // gfx1250 global_prefetch_b8 on next batch's plane
            __builtin_prefetch(Ip + (size_t)((b + 1) * 3) * PRJH * PRJW + o00, 0, 1);

        float s0 = s[(bc + 0) * HWc + pix];
        float s1 = s[(bc + 1) * HWc + pix];
        float s2 = s[(bc + 2) * HWc + pix];
        float smean = (s0 + s1 + s2) * (1.0f / 3.0f);
        float sv[3] = {s0, s1, s2};
#pragma unroll
        for (int c = 0; c < 3; ++c) {
            const float* plane = Ip + (size_t)(bc + c) * PRJH * PRJW;
            float val = f00 * plane[o00] + f01 * plane[o01] +
                        f10 * plane[o10] + f11 * plane[o11];
            float ipw = val * dl;
            int oidx = (bc + c) * HWc + pix;
            o_ipw[oidx]  = ipw;
            o_diff[oidx] = clamp01(ndl * ipw * sv[c]);
            o_spec[oidx] = clamp01(rdv * ipw * smean);
        }
    }
}

// ---------------------------------------------------------------------------
// Launch
// ---------------------------------------------------------------------------
extern "C" void kernel_launch(void* const* d_in, const int* in_sizes, int n_in,
                              void* d_out, int out_size, void* d_ws, size_t ws_size,
                              hipStream_t stream) {
    const float* Ip     = (const float*)d_in[0];
    const float* s      = (const float*)d_in[1];
    const float* depth  = (const float*)d_in[2];
    const float* npx    = (const float*)d_in[3];
    const float* RT     = (const float*)d_in[4];
    const float* Km     = (const float*)d_in[5];
    const float* org    = (const float*)d_in[6];
    const float* rect   = (const float*)d_in[7];
    const float* unrect = (const float*)d_in[8];

    float* out = (float*)d_out;
    const size_t HW = (size_t)HWc;
    float* o_ipw  = out;                        // (8,3,H,W)
    float* o_diff = out + 1 * NB * 3 * HW;      // (8,3,H,W)
    float* o_spec = out + 2 * NB * 3 * HW;      // (8,3,H,W)
    float* o_n    = out + 3 * NB * 3 * HW;      // (8,3,H,W)
    float* o_grid = out + 4 * NB * 3 * HW;      // (1,H,W,2)
    float* o_dl   = o_grid + 2 * HW;            // (1,1,H,W)

    float* ws    = (float*)d_ws;
    float* xyz   = ws;             // (3,H,W)
    float* xyzr  = ws + 3 * HW;    // (3,H,W)
    float* maskr = ws + 6 * HW;    // (H,W)
    float* nbuf  = ws + 7 * HW;    // (3,H,W)

    const int blocks = HWc / 256;  // 3072

    geom_kernel<<<blocks, 256, 0, stream>>>(depth, npx, RT, Km, o_grid, xyz);
    rect_sample_kernel<<<blocks, 256, 0, stream>>>(xyz, rect, xyzr);
    sort_mask_kernel<<<CAMH, 1024, 0, stream>>>(xyzr, RT, maskr);
    unrect_kernel<<<blocks, 256, 0, stream>>>(maskr, unrect, o_grid, o_dl);
    normals_kernel<<<blocks, 256, 0, stream>>>(depth, npx, nbuf, o_n);
    shade_kernel<<<blocks, 256, 0, stream>>>(Ip, s, depth, npx, org,
                                             o_grid, o_dl, nbuf,
                                             o_ipw, o_diff, o_spec);
}